// MambaDecoupledEKF_51539607552090
// MI455X (gfx1250) — compile-verified
//
#include <hip/hip_runtime.h>
#include <hip/hip_bf16.h>
#include <math.h>

typedef float v2f __attribute__((ext_vector_type(2)));
typedef float v8f __attribute__((ext_vector_type(8)));

static constexpr float DT       = 0.1f;
static constexpr float HDT2     = 0.5f * 0.1f * 0.1f;   // 0.5*DT*DT
static constexpr float EPS_PSD  = 1e-5f;
static constexpr float TWO_PI_F = 6.28318530717958647692f;
static constexpr float INV_2PI  = 0.15915494309189533577f;

__device__ __forceinline__ float softplusf(float x) {
    // logaddexp(x, 0) = max(x,0) + log1p(exp(-|x|))
    return fmaxf(x, 0.0f) + log1pf(expf(-fabsf(x)));
}

__device__ __forceinline__ float wrap_pi(float a) {
    // a - 2*pi*round(a/(2*pi)); rintf = round-half-even like jnp.round
    return fmaf(-TWO_PI_F, rintf(a * INV_2PI), a);
}

// Branchless F_POS element (constant-acceleration transition), m = row, k = col.
// k is always < 8; rows m >= 8 naturally evaluate to 0 (no comparisons match).
__device__ __forceinline__ float f_el(int m, int k) {
    return (float)(k == m)
         + DT   * (float)(k == m + 3)
         + HDT2 * (float)(k == m + 6);
}

__global__ __launch_bounds__(256)
void ekf_fused_kernel(const float* __restrict__ x_pos,
                      const float* __restrict__ P_pos,
                      const float* __restrict__ z_pos,
                      const float* __restrict__ Q_pos_raw,
                      const float* __restrict__ R_pos_raw,
                      const float* __restrict__ x_ori,
                      const float* __restrict__ P_ori,
                      const float* __restrict__ z_ori,
                      const float* __restrict__ Q_ori_raw,
                      const float* __restrict__ R_ori_raw,
                      float* __restrict__ out,
                      int n)
{
    const int lane  = threadIdx.x & 31;
    const int e     = blockIdx.x * 256 + threadIdx.x;
    const int wbase = e - lane;                 // first element of this wave
    const bool active = (e < n);

    // ---------------- predicted x_pos = F @ x ----------------
    float xp[8];

#if __has_builtin(__builtin_amdgcn_wmma_f32_16x16x4_f32)
    if (wbase + 31 < n) {
        // Batched F@x via V_WMMA_F32_16X16X4_F32:
        //   A[16x4] = F columns (shared constant, rows 8..15 zero)
        //   B[4x16] = x-vectors of 16 consecutive elements (one per column)
        //   C[16x16]: vgpr j, lane nn (<16) = (F x_{elem nn})[j]
        // Layouts per CDNA5 ISA 7.12.2: lane L<16 holds K={k0,k0+1}, lane L+16 K={k0+2,k0+3}.
        const int col = lane & 15;
        const int hi2 = (lane >> 4) << 1;       // 0 or 2

        // A operands for both K-chunks, branchless, computed once per lane.
        v2f a0, a1;
        a0.x = f_el(col, hi2);     a0.y = f_el(col, hi2 + 1);
        a1.x = f_el(col, hi2 + 4); a1.y = f_el(col, hi2 + 5);

        // B operands: all four b64 loads issued up front (one wait covers all).
        const float* xb0 = x_pos + (size_t)(wbase + col) * 8 + hi2;        // half 0
        const float* xb1 = x_pos + (size_t)(wbase + 16 + col) * 8 + hi2;   // half 1
        const v2f b00 = *(const v2f*)(xb0);       // k0 = 0
        const v2f b01 = *(const v2f*)(xb0 + 4);   // k0 = 4
        const v2f b10 = *(const v2f*)(xb1);
        const v2f b11 = *(const v2f*)(xb1 + 4);

        const v8f zero = {0.f, 0.f, 0.f, 0.f, 0.f, 0.f, 0.f, 0.f};
        v8f c0, c1;
        c0 = __builtin_amdgcn_wmma_f32_16x16x4_f32(false, a0, false, b00, (short)0, zero, false, false);
        c0 = __builtin_amdgcn_wmma_f32_16x16x4_f32(false, a1, false, b01, (short)0, c0,   false, false);
        c1 = __builtin_amdgcn_wmma_f32_16x16x4_f32(false, a0, false, b10, (short)0, zero, false, false);
        c1 = __builtin_amdgcn_wmma_f32_16x16x4_f32(false, a1, false, b11, (short)0, c1,   false, false);

        // Hand results back to owning lanes: element (wbase+lane) lives at
        // lane (lane&15) of c[lane>>4].
        const int src = (lane & 15) * 4;        // byte index for bpermute
#pragma unroll
        for (int j = 0; j < 8; ++j) {
            const int t0 = __builtin_amdgcn_ds_bpermute(src, __float_as_int(c0[j]));
            const int t1 = __builtin_amdgcn_ds_bpermute(src, __float_as_int(c1[j]));
            xp[j] = __int_as_float(lane < 16 ? t0 : t1);
        }
    } else if (active)
#else
    if (active)
#endif
    {
        // scalar fallback (partial tail wave)
        const float* xb = x_pos + (size_t)e * 8;
        float xv[8];
#pragma unroll
        for (int i = 0; i < 8; ++i) xv[i] = xb[i];
        xp[0] = xv[0] + DT * xv[3] + HDT2 * xv[6];
        xp[1] = xv[1] + DT * xv[4] + HDT2 * xv[7];
        xp[2] = xv[2] + DT * xv[5];
        xp[3] = xv[3] + DT * xv[6];
        xp[4] = xv[4] + DT * xv[7];
        xp[5] = xv[5]; xp[6] = xv[6]; xp[7] = xv[7];
    }
    if (!active) return;

    // ---------------- position filter (D=8, O=3) ----------------
    float P[8][8];
    {
        const float4* Pb = (const float4*)(P_pos + (size_t)e * 64);
#pragma unroll
        for (int i = 0; i < 8; ++i) {
            float4 lo = Pb[2 * i], hh = Pb[2 * i + 1];
            P[i][0] = lo.x; P[i][1] = lo.y; P[i][2] = lo.z; P[i][3] = lo.w;
            P[i][4] = hh.x; P[i][5] = hh.y; P[i][6] = hh.z; P[i][7] = hh.w;
        }
    }

    // Q = cholesky_to_psd(Q_pos_raw): only lower triangle of raw is used.
    float Q[8][8];
    {
        const float* Qb = Q_pos_raw + (size_t)e * 64;
        float L[8][8];
#pragma unroll
        for (int i = 0; i < 8; ++i) {
#pragma unroll
            for (int j = 0; j <= i; ++j) L[i][j] = Qb[i * 8 + j];
            L[i][i] = softplusf(L[i][i]) + EPS_PSD;
        }
#pragma unroll
        for (int i = 0; i < 8; ++i) {
#pragma unroll
            for (int j = 0; j <= i; ++j) {
                float s = 0.0f;
#pragma unroll
                for (int k = 0; k <= j; ++k) s = fmaf(L[i][k], L[j][k], s);
                Q[i][j] = s; Q[j][i] = s;
            }
        }
    }

    // P = F P F^T + Q, exploiting F = I + DT*S + 0.5DT^2*S^2 sparsity (in place).
#pragma unroll
    for (int c = 0; c < 8; ++c) {              // A = F P  (rows)
        P[0][c] = fmaf(HDT2, P[6][c], fmaf(DT, P[3][c], P[0][c]));
        P[1][c] = fmaf(HDT2, P[7][c], fmaf(DT, P[4][c], P[1][c]));
        P[2][c] = fmaf(DT, P[5][c], P[2][c]);
        P[3][c] = fmaf(DT, P[6][c], P[3][c]);
        P[4][c] = fmaf(DT, P[7][c], P[4][c]);
    }
#pragma unroll
    for (int r = 0; r < 8; ++r) {              // (A) F^T  (cols)
        P[r][0] = fmaf(HDT2, P[r][6], fmaf(DT, P[r][3], P[r][0]));
        P[r][1] = fmaf(HDT2, P[r][7], fmaf(DT, P[r][4], P[r][1]));
        P[r][2] = fmaf(DT, P[r][5], P[r][2]);
        P[r][3] = fmaf(DT, P[r][6], P[r][3]);
        P[r][4] = fmaf(DT, P[r][7], P[r][4]);
    }
#pragma unroll
    for (int r = 0; r < 8; ++r)
#pragma unroll
        for (int c = 0; c < 8; ++c) P[r][c] += Q[r][c];

    // S = P[0:3,0:3] + R;  Sinv via adjugate
    float Si[3][3];
    {
        const float* Rb = R_pos_raw + (size_t)e * 9;
        float l00 = softplusf(Rb[0]) + EPS_PSD;
        float l10 = Rb[3];
        float l11 = softplusf(Rb[4]) + EPS_PSD;
        float l20 = Rb[6], l21 = Rb[7];
        float l22 = softplusf(Rb[8]) + EPS_PSD;
        float R00 = l00 * l00;
        float R10 = l10 * l00;
        float R11 = fmaf(l10, l10, l11 * l11);
        float R20 = l20 * l00;
        float R21 = fmaf(l20, l10, l21 * l11);
        float R22 = fmaf(l20, l20, fmaf(l21, l21, l22 * l22));
        float S00 = P[0][0] + R00, S01 = P[0][1] + R10, S02 = P[0][2] + R20;
        float S10 = P[1][0] + R10, S11 = P[1][1] + R11, S12 = P[1][2] + R21;
        float S20 = P[2][0] + R20, S21 = P[2][1] + R21, S22 = P[2][2] + R22;
        float a00 = S11 * S22 - S12 * S21;
        float a01 = S02 * S21 - S01 * S22;
        float a02 = S01 * S12 - S02 * S11;
        float a10 = S12 * S20 - S10 * S22;
        float a11 = S00 * S22 - S02 * S20;
        float a12 = S02 * S10 - S00 * S12;
        float a20 = S10 * S21 - S11 * S20;
        float a21 = S01 * S20 - S00 * S21;
        float a22 = S00 * S11 - S01 * S10;
        float det = fmaf(S00, a00, fmaf(S01, a10, S02 * a20));
        float id  = 1.0f / det;
        Si[0][0] = a00 * id; Si[0][1] = a01 * id; Si[0][2] = a02 * id;
        Si[1][0] = a10 * id; Si[1][1] = a11 * id; Si[1][2] = a12 * id;
        Si[2][0] = a20 * id; Si[2][1] = a21 * id; Si[2][2] = a22 * id;
    }

    // K = P H^T Sinv  (H^T selects columns 0..2 of P)
    float K[8][3];
#pragma unroll
    for (int r = 0; r < 8; ++r)
#pragma unroll
        for (int c = 0; c < 3; ++c)
            K[r][c] = fmaf(P[r][0], Si[0][c], fmaf(P[r][1], Si[1][c], P[r][2] * Si[2][c]));

    const float* zb = z_pos + (size_t)e * 3;
    float in0 = zb[0] - xp[0], in1 = zb[1] - xp[1], in2 = zb[2] - xp[2];
    float xn[8];
#pragma unroll
    for (int r = 0; r < 8; ++r)
        xn[r] = xp[r] + fmaf(K[r][0], in0, fmaf(K[r][1], in1, K[r][2] * in2));

    // P = (I - K H) P = P - K @ P[0:3,:]
    {
        float r0[8], r1[8], r2[8];
#pragma unroll
        for (int c = 0; c < 8; ++c) { r0[c] = P[0][c]; r1[c] = P[1][c]; r2[c] = P[2][c]; }
#pragma unroll
        for (int r = 0; r < 8; ++r)
#pragma unroll
            for (int c = 0; c < 8; ++c)
                P[r][c] -= fmaf(K[r][0], r0[c], fmaf(K[r][1], r1[c], K[r][2] * r2[c]));
    }

    // ---------------- orientation filter (D=2, O=1, wrapped) ----------------
    float xo0f, xo1f, Pn00, Pn01, Pn10, Pn11;
    {
        const float2 xo = ((const float2*)x_ori)[e];
        const float4 Po = ((const float4*)P_ori)[e];       // P00 P01 P10 P11
        const float  zo = z_ori[e];
        const float4 Qr = ((const float4*)Q_ori_raw)[e];   // raw 2x2
        const float  rr0 = R_ori_raw[e];

        float d0   = softplusf(Qr.x) + EPS_PSD;
        float lo10 = Qr.z;
        float d1   = softplusf(Qr.w) + EPS_PSD;
        float Qo00 = d0 * d0;
        float Qo01 = d0 * lo10;
        float Qo11 = fmaf(lo10, lo10, d1 * d1);
        float rs = softplusf(rr0) + EPS_PSD;
        float Ro = rs * rs;

        float x0 = wrap_pi(fmaf(DT, xo.y, xo.x));
        float x1 = xo.y;
        // P' = F P F^T + Q with F=[[1,DT],[0,1]]
        float A00 = fmaf(DT, Po.z, Po.x);
        float A01 = fmaf(DT, Po.w, Po.y);
        float Pp00 = fmaf(DT, A01, A00) + Qo00;
        float Pp01 = A01 + Qo01;
        float Pp10 = fmaf(DT, Po.w, Po.z) + Qo01;
        float Pp11 = Po.w + Qo11;

        float S  = Pp00 + Ro;
        float iS = 1.0f / S;
        float K0 = Pp00 * iS, K1 = Pp10 * iS;
        float inn = wrap_pi(zo - x0);
        xo0f = wrap_pi(fmaf(K0, inn, x0));
        xo1f = fmaf(K1, inn, x1);
        Pn00 = (1.0f - K0) * Pp00;
        Pn01 = (1.0f - K0) * Pp01;
        Pn10 = Pp10 - K1 * Pp00;
        Pn11 = Pp11 - K1 * Pp01;
    }

    // ---------------- output: [x_pos(8) | P_pos(64) | x_ori(2) | P_ori(4)] ----------------
    float* o = out + (size_t)e * 78;           // base is 8B aligned -> use b64 stores
#pragma unroll
    for (int i = 0; i < 4; ++i) { float2 t; t.x = xn[2 * i]; t.y = xn[2 * i + 1]; ((float2*)o)[i] = t; }
#pragma unroll
    for (int r = 0; r < 8; ++r)
#pragma unroll
        for (int i = 0; i < 4; ++i) {
            float2 t; t.x = P[r][2 * i]; t.y = P[r][2 * i + 1];
            *(float2*)(o + 8 + r * 8 + 2 * i) = t;
        }
    { float2 t; t.x = xo0f; t.y = xo1f; *(float2*)(o + 72) = t; }
    { float2 t; t.x = Pn00; t.y = Pn01; *(float2*)(o + 74) = t; }
    { float2 t; t.x = Pn10; t.y = Pn11; *(float2*)(o + 76) = t; }
}

extern "C" void kernel_launch(void* const* d_in, const int* in_sizes, int n_in,
                              void* d_out, int out_size, void* d_ws, size_t ws_size,
                              hipStream_t stream) {
    const float* x_pos     = (const float*)d_in[0];
    const float* P_pos     = (const float*)d_in[1];
    const float* z_pos     = (const float*)d_in[2];
    const float* Q_pos_raw = (const float*)d_in[3];
    const float* R_pos_raw = (const float*)d_in[4];
    const float* x_ori     = (const float*)d_in[5];
    const float* P_ori     = (const float*)d_in[6];
    const float* z_ori     = (const float*)d_in[7];
    const float* Q_ori_raw = (const float*)d_in[8];
    const float* R_ori_raw = (const float*)d_in[9];
    float* out = (float*)d_out;

    const int n = in_sizes[0] / 8;             // batch size (x_pos is [B,8,1])
    const int grid = (n + 255) / 256;
    ekf_fused_kernel<<<grid, 256, 0, stream>>>(x_pos, P_pos, z_pos, Q_pos_raw, R_pos_raw,
                                               x_ori, P_ori, z_ori, Q_ori_raw, R_ori_raw,
                                               out, n);
}